// MoELoRALayer_8839042695777
// MI455X (gfx1250) — compile-verified
//
#include <hip/hip_runtime.h>
#include <hip/hip_bf16.h>

// ---------------- problem constants (match reference) ----------------
#define NTOK   2048
#define HID    1024
#define INTER  2816
#define NEXP   8
#define NADP   2
#define RMAX   16
#define TOPK   2
#define NSEQ   8
#define NPAIR  (NTOK * TOPK)   // 4096

typedef float v2f __attribute__((ext_vector_type(2)));
typedef float v8f __attribute__((ext_vector_type(8)));

// ---------------------------------------------------------------------
// 0) zero expert counters
// ---------------------------------------------------------------------
__global__ void zero_counts_kernel(int* counts) {
    if (threadIdx.x < NEXP) counts[threadIdx.x] = 0;
}

// ---------------------------------------------------------------------
// 1) routing: adapter/scale per token + per-expert pair lists
// ---------------------------------------------------------------------
__global__ void routing_kernel(const int* __restrict__ topk_ids,
                               const int* __restrict__ weight_indices,
                               const int* __restrict__ seq_lens,
                               const float* __restrict__ scalings,
                               int* __restrict__ counts,
                               int* __restrict__ lists,
                               int* __restrict__ adapter,
                               float* __restrict__ scale_tok) {
    int n = blockIdx.x * 256 + threadIdx.x;
    if (n >= NTOK) return;
    int cum = 0, seq = NSEQ - 1;
    for (int s = 0; s < NSEQ; ++s) {
        cum += seq_lens[s];
        if (n < cum) { seq = s; break; }
    }
    int a = weight_indices[seq];
    adapter[n]   = a;
    scale_tok[n] = scalings[a];
    for (int k = 0; k < TOPK; ++k) {
        int e   = topk_ids[n * TOPK + k];
        int pos = atomicAdd(&counts[e], 1);
        lists[e * NPAIR + pos] = n * TOPK + k;
    }
}

// ---------------------------------------------------------------------
// 2) z_gate / z_up : z[n,e,r] = sum_h x[n,h] * A[a,e,r,h]  (rank-masked)
// ---------------------------------------------------------------------
__global__ __launch_bounds__(256)
void lora_z_kernel(const float* __restrict__ x,
                   const float* __restrict__ ga,
                   const float* __restrict__ ua,
                   const int* __restrict__ adapter,
                   const int* __restrict__ ranks,
                   float* __restrict__ zg,
                   float* __restrict__ zu) {
    __shared__ float xs[8][HID];
    __shared__ int sAd[8];
    __shared__ int sRk[8];
    int nb = blockIdx.x * 8;
    int t  = threadIdx.x;
    for (int i = t; i < 8 * HID; i += 256) {
        int j = i >> 10;
        xs[j][i & (HID - 1)] = x[(size_t)(nb + j) * HID + (i & (HID - 1))];
    }
    if (t < 8) { int a = adapter[nb + t]; sAd[t] = a; sRk[t] = ranks[a]; }
    __syncthreads();

    int ad[8], rk[8];
#pragma unroll
    for (int j = 0; j < 8; ++j) { ad[j] = sAd[j]; rk[j] = sRk[j]; }

    int which = t >> 7;          // 0 = gate, 1 = up
    int e = (t >> 4) & 7;
    int r = t & 15;
    const float* base = which ? ua : ga;
    const float* row0 = base + (((size_t)0 * NEXP + e) * RMAX + r) * HID;
    const float* row1 = base + (((size_t)1 * NEXP + e) * RMAX + r) * HID;

    float acc[8];
#pragma unroll
    for (int j = 0; j < 8; ++j) acc[j] = 0.0f;

    for (int h = 0; h < HID; ++h) {
        float f0 = row0[h], f1 = row1[h];
#pragma unroll
        for (int j = 0; j < 8; ++j) acc[j] += xs[j][h] * (ad[j] ? f1 : f0);
    }
    float* dst = which ? zu : zg;
#pragma unroll
    for (int j = 0; j < 8; ++j) {
        float v = (r < rk[j]) ? acc[j] : 0.0f;
        dst[((size_t)(nb + j) * NEXP + e) * RMAX + r] = v;
    }
}

// ---------------------------------------------------------------------
// 3) gate_up GEMM (fp32 WMMA) + fused LoRA-B + silu*up epilogue
//    tile: 32 pairs x 128 inter-cols; each wave: 4 accumulators
//    (2 row-fragments x {gate,up}) sharing every B fragment.
//    Register-staged prefetch of the next K-chunk overlaps global
//    latency with the 32-WMMA compute burst.
// ---------------------------------------------------------------------
__global__ __launch_bounds__(256)
void gateup_kernel(const float* __restrict__ x,
                   const float* __restrict__ Wgu,
                   const int* __restrict__ lists,
                   const int* __restrict__ counts,
                   const int* __restrict__ adapter,
                   const float* __restrict__ scale_tok,
                   const float* __restrict__ zg,
                   const float* __restrict__ zu,
                   const float* __restrict__ gb,
                   const float* __restrict__ ub,
                   float* __restrict__ act) {
    const int e   = blockIdx.z;
    const int mt  = blockIdx.y;
    const int cnt = counts[e];
    if (mt * 32 >= cnt) return;
    const int colBase = blockIdx.x * 128;

    __shared__ float As[32][33];
    __shared__ float Bg[128][33];
    __shared__ float Bu[128][33];
    __shared__ int   sPid[32];
    __shared__ int   sTok[32];
    __shared__ int   sAd[32];
    __shared__ int   sValid[32];
    __shared__ float sScale[32];

    const int t = threadIdx.x;
    if (t < 32) {
        int idx   = mt * 32 + t;
        int valid = (idx < cnt) ? 1 : 0;
        int pid   = lists[e * NPAIR + (valid ? idx : mt * 32)];
        sPid[t] = pid; sValid[t] = valid;
        int n = pid >> 1;
        sTok[t] = n; sAd[t] = adapter[n]; sScale[t] = scale_tok[n];
    }
    __syncthreads();

    const int wave = t >> 5, lane = t & 31, half = lane >> 4, lq = lane & 15;
    const int kk = t & 31, ar = t >> 5;          // A staging: row base / col
    const int kq = (t & 7) * 4, cB = t >> 3;     // B staging: col base / k
    v8f cg0 = {}, cg1 = {}, cu0 = {}, cu1 = {};
    const float* Wg = Wgu + (size_t)e * (2 * INTER) * HID;

    // ---- preload K-chunk 0 into LDS ----
    {
#pragma unroll
        for (int q = 0; q < 4; ++q)
            As[ar + 8 * q][kk] = x[(size_t)sTok[ar + 8 * q] * HID + kk];
#pragma unroll
        for (int q = 0; q < 4; ++q) {
            int c = cB + 32 * q;
            const float4 vg = *(const float4*)(Wg + (size_t)(colBase + c) * HID + kq);
            Bg[c][kq] = vg.x; Bg[c][kq + 1] = vg.y; Bg[c][kq + 2] = vg.z; Bg[c][kq + 3] = vg.w;
            const float4 vu = *(const float4*)(Wg + (size_t)(INTER + colBase + c) * HID + kq);
            Bu[c][kq] = vu.x; Bu[c][kq + 1] = vu.y; Bu[c][kq + 2] = vu.z; Bu[c][kq + 3] = vu.w;
        }
    }
    __syncthreads();

    const int NCHUNK = HID / 32;
    for (int chunk = 0; chunk < NCHUNK; ++chunk) {
        const bool pf = (chunk + 1 < NCHUNK);
        const int k0n = (chunk + 1) * 32;
        float  aReg[4];
        float4 bgReg[4], buReg[4];
        if (pf) {   // issue next-chunk global loads before compute
#pragma unroll
            for (int q = 0; q < 4; ++q)
                aReg[q] = x[(size_t)sTok[ar + 8 * q] * HID + k0n + kk];
#pragma unroll
            for (int q = 0; q < 4; ++q) {
                int c = cB + 32 * q;
                bgReg[q] = *(const float4*)(Wg + (size_t)(colBase + c) * HID + k0n + kq);
                buReg[q] = *(const float4*)(Wg + (size_t)(INTER + colBase + c) * HID + k0n + kq);
            }
        }
#pragma unroll
        for (int ks = 0; ks < 32; ks += 4) {
            v2f a0, a1, bg, bu;
            a0.x = As[lq][ks + 2 * half];      a0.y = As[lq][ks + 2 * half + 1];
            a1.x = As[16 + lq][ks + 2 * half]; a1.y = As[16 + lq][ks + 2 * half + 1];
            int c = wave * 16 + lq;
            bg.x = Bg[c][ks + 2 * half];       bg.y = Bg[c][ks + 2 * half + 1];
            bu.x = Bu[c][ks + 2 * half];       bu.y = Bu[c][ks + 2 * half + 1];
            cg0 = __builtin_amdgcn_wmma_f32_16x16x4_f32(false, a0, false, bg, (short)0, cg0, false, false);
            cg1 = __builtin_amdgcn_wmma_f32_16x16x4_f32(false, a1, false, bg, (short)0, cg1, false, false);
            cu0 = __builtin_amdgcn_wmma_f32_16x16x4_f32(false, a0, false, bu, (short)0, cu0, false, false);
            cu1 = __builtin_amdgcn_wmma_f32_16x16x4_f32(false, a1, false, bu, (short)0, cu1, false, false);
        }
        __syncthreads();
        if (pf) {
#pragma unroll
            for (int q = 0; q < 4; ++q) As[ar + 8 * q][kk] = aReg[q];
#pragma unroll
            for (int q = 0; q < 4; ++q) {
                int c = cB + 32 * q;
                Bg[c][kq] = bgReg[q].x; Bg[c][kq + 1] = bgReg[q].y; Bg[c][kq + 2] = bgReg[q].z; Bg[c][kq + 3] = bgReg[q].w;
                Bu[c][kq] = buReg[q].x; Bu[c][kq + 1] = buReg[q].y; Bu[c][kq + 2] = buReg[q].z; Bu[c][kq + 3] = buReg[q].w;
            }
            __syncthreads();
        }
    }

    // epilogue: add scaled LoRA-B, silu(gate)*up, write act rows
    const int col = colBase + wave * 16 + lq;
#pragma unroll
    for (int jj = 0; jj < 16; ++jj) {
        int j = jj & 7;
        int m = j + 8 * half + ((jj >> 3) << 4);   // rows 0-15 then 16-31
        if (!sValid[m]) continue;
        float cgv = (jj < 8) ? cg0[j] : cg1[j];
        float cuv = (jj < 8) ? cu0[j] : cu1[j];
        int n = sTok[m]; int a = sAd[m]; float s = sScale[m];
        const float* zgr = zg + ((size_t)n * NEXP + e) * RMAX;
        const float* zur = zu + ((size_t)n * NEXP + e) * RMAX;
        const float* gbr = gb + (((size_t)a * NEXP + e) * INTER + col) * RMAX;
        const float* ubr = ub + (((size_t)a * NEXP + e) * INTER + col) * RMAX;
        float lg = 0.0f, lu = 0.0f;
#pragma unroll
        for (int r = 0; r < RMAX; ++r) { lg += zgr[r] * gbr[r]; lu += zur[r] * ubr[r]; }
        float g = cgv + s * lg;
        float u = cuv + s * lu;
        float av = (g / (1.0f + __expf(-g))) * u;
        act[(size_t)sPid[m] * INTER + col] = av;
    }
}

// ---------------------------------------------------------------------
// 4) zd[pair,r] = sum_i act[pair,i] * da[a,e,r,i]  (rank-masked)
// ---------------------------------------------------------------------
__global__ __launch_bounds__(256)
void zd_kernel(const float* __restrict__ act,
               const float* __restrict__ da,
               const int* __restrict__ topk_ids,
               const int* __restrict__ adapter,
               const int* __restrict__ ranks,
               float* __restrict__ zd) {
    int pid = blockIdx.x;
    int n = pid >> 1;
    int e = topk_ids[pid];
    int a = adapter[n];
    __shared__ float as[INTER];
    __shared__ float red[256];
    int t = threadIdx.x;
    for (int i = t; i < INTER; i += 256) as[i] = act[(size_t)pid * INTER + i];
    __syncthreads();
    int r = t >> 4, sub = t & 15;
    const float* row = da + (((size_t)a * NEXP + e) * RMAX + r) * INTER;
    float p = 0.0f;
    for (int i = sub; i < INTER; i += 16) p += as[i] * row[i];
    red[t] = p;
    __syncthreads();
    if (t < RMAX) {
        float s = 0.0f;
#pragma unroll
        for (int q = 0; q < 16; ++q) s += red[t * 16 + q];
        zd[(size_t)pid * RMAX + t] = (t < ranks[a]) ? s : 0.0f;
    }
}

// ---------------------------------------------------------------------
// 5) down GEMM (fp32 WMMA) + fused LoRA-down + routing-weight epilogue
//    tile: 32 pairs x 128 hidden-cols; same structure as gate_up.
// ---------------------------------------------------------------------
__global__ __launch_bounds__(256)
void down_kernel(const float* __restrict__ act,
                 const float* __restrict__ Wd,
                 const int* __restrict__ lists,
                 const int* __restrict__ counts,
                 const int* __restrict__ adapter,
                 const float* __restrict__ scale_tok,
                 const float* __restrict__ tw,
                 const float* __restrict__ zd,
                 const float* __restrict__ db,
                 float* __restrict__ out_partial) {
    const int e   = blockIdx.z;
    const int mt  = blockIdx.y;
    const int cnt = counts[e];
    if (mt * 32 >= cnt) return;
    const int colBase = blockIdx.x * 128;

    __shared__ float As[32][33];
    __shared__ float Bs[128][33];
    __shared__ int   sPid[32];
    __shared__ int   sAd[32];
    __shared__ int   sValid[32];
    __shared__ float sScale[32];
    __shared__ float sW[32];

    const int t = threadIdx.x;
    if (t < 32) {
        int idx   = mt * 32 + t;
        int valid = (idx < cnt) ? 1 : 0;
        int pid   = lists[e * NPAIR + (valid ? idx : mt * 32)];
        sPid[t] = pid; sValid[t] = valid;
        int n = pid >> 1;
        sAd[t] = adapter[n]; sScale[t] = scale_tok[n]; sW[t] = tw[pid];
    }
    __syncthreads();

    const int wave = t >> 5, lane = t & 31, half = lane >> 4, lq = lane & 15;
    const int kk = t & 31, ar = t >> 5;
    const int kq = (t & 7) * 4, cB = t >> 3;
    v8f c0 = {}, c1 = {};
    const float* W = Wd + (size_t)e * HID * INTER;

    {
#pragma unroll
        for (int q = 0; q < 4; ++q)
            As[ar + 8 * q][kk] = act[(size_t)sPid[ar + 8 * q] * INTER + kk];
#pragma unroll
        for (int q = 0; q < 4; ++q) {
            int c = cB + 32 * q;
            const float4 v = *(const float4*)(W + (size_t)(colBase + c) * INTER + kq);
            Bs[c][kq] = v.x; Bs[c][kq + 1] = v.y; Bs[c][kq + 2] = v.z; Bs[c][kq + 3] = v.w;
        }
    }
    __syncthreads();

    const int NCHUNK = INTER / 32;
    for (int chunk = 0; chunk < NCHUNK; ++chunk) {
        const bool pf = (chunk + 1 < NCHUNK);
        const int k0n = (chunk + 1) * 32;
        float  aReg[4];
        float4 bReg[4];
        if (pf) {
#pragma unroll
            for (int q = 0; q < 4; ++q)
                aReg[q] = act[(size_t)sPid[ar + 8 * q] * INTER + k0n + kk];
#pragma unroll
            for (int q = 0; q < 4; ++q) {
                int c = cB + 32 * q;
                bReg[q] = *(const float4*)(W + (size_t)(colBase + c) * INTER + k0n + kq);
            }
        }
#pragma unroll
        for (int ks = 0; ks < 32; ks += 4) {
            v2f a0, a1, b;
            a0.x = As[lq][ks + 2 * half];      a0.y = As[lq][ks + 2 * half + 1];
            a1.x = As[16 + lq][ks + 2 * half]; a1.y = As[16 + lq][ks + 2 * half + 1];
            int c = wave * 16 + lq;
            b.x = Bs[c][ks + 2 * half];        b.y = Bs[c][ks + 2 * half + 1];
            c0 = __builtin_amdgcn_wmma_f32_16x16x4_f32(false, a0, false, b, (short)0, c0, false, false);
            c1 = __builtin_amdgcn_wmma_f32_16x16x4_f32(false, a1, false, b, (short)0, c1, false, false);
        }
        __syncthreads();
        if (pf) {
#pragma unroll
            for (int q = 0; q < 4; ++q) As[ar + 8 * q][kk] = aReg[q];
#pragma unroll
            for (int q = 0; q < 4; ++q) {
                int c = cB + 32 * q;
                Bs[c][kq] = bReg[q].x; Bs[c][kq + 1] = bReg[q].y; Bs[c][kq + 2] = bReg[q].z; Bs[c][kq + 3] = bReg[q].w;
            }
            __syncthreads();
        }
    }

    const int h = colBase + wave * 16 + lq;
#pragma unroll
    for (int jj = 0; jj < 16; ++jj) {
        int j = jj & 7;
        int m = j + 8 * half + ((jj >> 3) << 4);
        if (!sValid[m]) continue;
        float cv = (jj < 8) ? c0[j] : c1[j];
        int pid = sPid[m]; int a = sAd[m];
        const float* zr  = zd + (size_t)pid * RMAX;
        const float* dbr = db + (((size_t)a * NEXP + e) * HID + h) * RMAX;
        float ld = 0.0f;
#pragma unroll
        for (int r = 0; r < RMAX; ++r) ld += zr[r] * dbr[r];
        out_partial[(size_t)pid * HID + h] = sW[m] * (cv + sScale[m] * ld);
    }
}

// ---------------------------------------------------------------------
// 6) deterministic 2-way reduction over top-k slots
// ---------------------------------------------------------------------
__global__ void reduce_kernel(const float* __restrict__ op, float* __restrict__ out) {
    size_t i = (size_t)blockIdx.x * 256 + threadIdx.x;
    if (i >= (size_t)NTOK * HID) return;
    size_t n = i >> 10, h = i & (HID - 1);
    out[i] = op[(n * 2) * HID + h] + op[(n * 2 + 1) * HID + h];
}

// ---------------------------------------------------------------------
extern "C" void kernel_launch(void* const* d_in, const int* in_sizes, int n_in,
                              void* d_out, int out_size, void* d_ws, size_t ws_size,
                              hipStream_t stream) {
    const float* x         = (const float*)d_in[0];
    const int*   topk_ids  = (const int*)d_in[1];
    const float* tw        = (const float*)d_in[2];
    const float* ga        = (const float*)d_in[3];
    const float* gb        = (const float*)d_in[4];
    const float* ua        = (const float*)d_in[5];
    const float* ub        = (const float*)d_in[6];
    const float* da        = (const float*)d_in[7];
    const float* db        = (const float*)d_in[8];
    const int*   wi        = (const int*)d_in[9];
    const int*   sl        = (const int*)d_in[10];
    const int*   ranks     = (const int*)d_in[11];
    const float* scalings  = (const float*)d_in[12];
    const float* Wgu       = (const float*)d_in[13];
    const float* Wd        = (const float*)d_in[14];
    float*       out       = (float*)d_out;

    char* ws = (char*)d_ws;
    size_t off = 0;
    auto alloc = [&](size_t bytes) -> void* {
        void* p = ws + off;
        off = (off + bytes + 255) & ~(size_t)255;
        return p;
    };
    int*   counts      = (int*)  alloc(NEXP * sizeof(int));
    int*   lists       = (int*)  alloc((size_t)NEXP * NPAIR * sizeof(int));
    int*   adapter     = (int*)  alloc(NTOK * sizeof(int));
    float* scale_tok   = (float*)alloc(NTOK * sizeof(float));
    float* zg          = (float*)alloc((size_t)NTOK * NEXP * RMAX * sizeof(float));
    float* zu          = (float*)alloc((size_t)NTOK * NEXP * RMAX * sizeof(float));
    float* zd          = (float*)alloc((size_t)NPAIR * RMAX * sizeof(float));
    float* act         = (float*)alloc((size_t)NPAIR * INTER * sizeof(float));
    float* out_partial = (float*)alloc((size_t)NPAIR * HID * sizeof(float));

    zero_counts_kernel<<<1, 32, 0, stream>>>(counts);
    routing_kernel<<<(NTOK + 255) / 256, 256, 0, stream>>>(
        topk_ids, wi, sl, scalings, counts, lists, adapter, scale_tok);
    lora_z_kernel<<<NTOK / 8, 256, 0, stream>>>(x, ga, ua, adapter, ranks, zg, zu);
    gateup_kernel<<<dim3(INTER / 128, NPAIR / 32, NEXP), 256, 0, stream>>>(
        x, Wgu, lists, counts, adapter, scale_tok, zg, zu, gb, ub, act);
    zd_kernel<<<NPAIR, 256, 0, stream>>>(act, da, topk_ids, adapter, ranks, zd);
    down_kernel<<<dim3(HID / 128, NPAIR / 32, NEXP), 256, 0, stream>>>(
        act, Wd, lists, counts, adapter, scale_tok, tw, zd, db, out_partial);
    reduce_kernel<<<(NTOK * HID + 255) / 256, 256, 0, stream>>>(out_partial, out);
}